// BallQuery_42099269435597
// MI455X (gfx1250) — compile-verified
//
#include <hip/hip_runtime.h>

// Ball query (B=8, N=8192, M=2048, radius=0.2, nsample=32) for gfx1250.
//
// d2 = |q|^2 - 2 q.k + |k|^2 as a K=4 WMMA:
//   A[m]=(-2qx,-2qy,-2qz,1), B[:,n]=(kx,ky,kz,|k|^2), C[m][n]=|q_m|^2
//   => D = A*B + C = d2 tile (16x16) per V_WMMA_F32_16X16X4_F32.
// Selection on d2 directly: sqrt is monotone and
//   sqrt_rn(clip(d2,0)) <= 0.2f  <=>  d2 <= 0x1.47AE16p-5f  (exact float equivalence).
// One wave32 per (32 queries x 1 point-segment); 32 points per iteration (4 WMMAs),
// software-pipelined point loads, per-lane sorted top-32 in LDS.
// N split into S segments for occupancy (S*512 waves), S-way merge kernel after.

typedef __attribute__((ext_vector_type(2))) float v2f;
typedef __attribute__((ext_vector_type(8))) float v8f;

#define BQ_NSAMPLE 32
#define BQ_B       8
// Largest float d2 with sqrt_rn(d2) <= 0.2f  (bits 0x3D23D70B)
#define BQ_R2EQ    0x1.47AE16p-5f

#define BQ_BIT(v, b) (msk |= ((v) <= BQ_R2EQ) ? (1u << (b)) : 0u)

__global__ __launch_bounds__(32) void ball_query_seg_kernel(
    const float* __restrict__ xyz,      // (B, N, 3)
    const float* __restrict__ new_xyz,  // (B, M, 3)
    int*   __restrict__ out,            // (B, M, 32)   [S==1 path]
    float* __restrict__ wd,             // (S, 32, totalQ) sorted d2   [S>1]
    int*   __restrict__ wi,             // (S, 32, totalQ) indices     [S>1]
    int*   __restrict__ wc,             // (S, totalQ) in-radius count (capped 32)
    int N, int M, int S, int totalQ)
{
    __shared__ float qqs[32];
    __shared__ float tile[32 * 32];           // d2 tile: 32 query rows x 32 points
    __shared__ float dlist[BQ_NSAMPLE * 32];  // [entry][lane] sorted d2
    __shared__ int   ilist[BQ_NSAMPLE * 32];  // [entry][lane] point index

    const int lid  = threadIdx.x;      // 0..31 (wave32)
    const int half = lid >> 4;
    const int lm   = lid & 15;

    const int seg   = blockIdx.x % S;
    const int qblk  = blockIdx.x / S;
    const int qbase = qblk * 32;         // M%32==0 => all 32 queries in same batch
    const int b     = qbase / M;
    const int mq    = qbase - b * M;

    const float* Q = new_xyz + ((size_t)b * M + mq) * 3;
    const float* P = xyz + (size_t)b * N * 3;

    {   // |q|^2 of owned query -> LDS for C tiles
        float qx = Q[lid * 3 + 0], qy = Q[lid * 3 + 1], qz = Q[lid * 3 + 2];
        qqs[lid] = qx * qx + qy * qy + qz * qz;
    }

    // A tiles per ISA 16x4 f32 layout: lanes 0-15 hold {K0,K1}, lanes 16-31 {K2,K3}
    float q1x = Q[lm * 3 + 0],        q1y = Q[lm * 3 + 1],        q1z = Q[lm * 3 + 2];
    float q2x = Q[(16 + lm) * 3 + 0], q2y = Q[(16 + lm) * 3 + 1], q2z = Q[(16 + lm) * 3 + 2];
    v2f a1, a2;
    a1[0] = half ? (-2.0f * q1z) : (-2.0f * q1x);
    a1[1] = half ? 1.0f          : (-2.0f * q1y);
    a2[0] = half ? (-2.0f * q2z) : (-2.0f * q2x);
    a2[1] = half ? 1.0f          : (-2.0f * q2y);
    __syncthreads();

    // C tiles per 16x16 f32 C/D layout: VGPR r, lane-half h -> row r + 8h
    v8f c1, c2;
#pragma unroll
    for (int r = 0; r < 8; ++r) {
        c1[r] = qqs[r + 8 * half];
        c2[r] = qqs[16 + r + 8 * half];
    }

    int   cnt   = 0;
    float worst = 1e30f;

    const int segN    = N / S;
    const int nbeg    = seg * segN;
    const int chunks  = segN >> 5;          // 32 points per iteration
    const int rowbase = lid * 32;

    // Software pipeline: preload first chunk's two points per lane
    float px0, py0, pz0, px1, py1, pz1;
    {
        const size_t p0 = (size_t)(nbeg + lm) * 3, p1 = (size_t)(nbeg + 16 + lm) * 3;
        px0 = P[p0 + 0]; py0 = P[p0 + 1]; pz0 = P[p0 + 2];
        px1 = P[p1 + 0]; py1 = P[p1 + 1]; pz1 = P[p1 + 2];
    }

    for (int c = 0; c < chunks; ++c) {
        const int n0 = nbeg + (c << 5);
        float kk0 = px0 * px0 + py0 * py0 + pz0 * pz0;
        float kk1 = px1 * px1 + py1 * py1 + pz1 * pz1;
        v2f bm0, bm1;                   // B 4x16: lanes 0-15 rows K0,K1; 16-31 K2,K3
        bm0[0] = half ? pz0 : px0;  bm0[1] = half ? kk0 : py0;
        bm1[0] = half ? pz1 : px1;  bm1[1] = half ? kk1 : py1;

        // Issue next chunk's loads now (clamped on last iter); overlap with WMMA+scan
        {
            const int cn = (c + 1 < chunks) ? (c + 1) : c;
            const size_t np0 = (size_t)(nbeg + (cn << 5) + lm) * 3;
            const size_t np1 = np0 + 48;   // +16 points * 3
            px0 = P[np0 + 0]; py0 = P[np0 + 1]; pz0 = P[np0 + 2];
            px1 = P[np1 + 0]; py1 = P[np1 + 1]; pz1 = P[np1 + 2];
        }

        v8f d1 = __builtin_amdgcn_wmma_f32_16x16x4_f32(   // q 0-15  x p 0-15
            false, a1, false, bm0, (short)0, c1, false, false);
        v8f d2 = __builtin_amdgcn_wmma_f32_16x16x4_f32(   // q 16-31 x p 0-15
            false, a2, false, bm0, (short)0, c2, false, false);
        v8f d3 = __builtin_amdgcn_wmma_f32_16x16x4_f32(   // q 0-15  x p 16-31
            false, a1, false, bm1, (short)0, c1, false, false);
        v8f d4 = __builtin_amdgcn_wmma_f32_16x16x4_f32(   // q 16-31 x p 16-31
            false, a2, false, bm1, (short)0, c2, false, false);

#pragma unroll
        for (int r = 0; r < 8; ++r) {
            int mrow = r + 8 * half;
            tile[mrow * 32 + lm]             = d1[r];   // raw d2; clamp deferred
            tile[(16 + mrow) * 32 + lm]      = d2[r];
            tile[mrow * 32 + 16 + lm]        = d3[r];
            tile[(16 + mrow) * 32 + 16 + lm] = d4[r];
        }
        __syncthreads();

        // Lane lid owns query (qbase+lid): branchless 32-bit mask, sparse pop-loop
        {
            const float4* trow = reinterpret_cast<const float4*>(&tile[rowbase]);
            float4 t0 = trow[0], t1 = trow[1], t2 = trow[2], t3 = trow[3];
            float4 t4 = trow[4], t5 = trow[5], t6 = trow[6], t7 = trow[7];
            unsigned msk = 0;
            BQ_BIT(t0.x,  0); BQ_BIT(t0.y,  1); BQ_BIT(t0.z,  2); BQ_BIT(t0.w,  3);
            BQ_BIT(t1.x,  4); BQ_BIT(t1.y,  5); BQ_BIT(t1.z,  6); BQ_BIT(t1.w,  7);
            BQ_BIT(t2.x,  8); BQ_BIT(t2.y,  9); BQ_BIT(t2.z, 10); BQ_BIT(t2.w, 11);
            BQ_BIT(t3.x, 12); BQ_BIT(t3.y, 13); BQ_BIT(t3.z, 14); BQ_BIT(t3.w, 15);
            BQ_BIT(t4.x, 16); BQ_BIT(t4.y, 17); BQ_BIT(t4.z, 18); BQ_BIT(t4.w, 19);
            BQ_BIT(t5.x, 20); BQ_BIT(t5.y, 21); BQ_BIT(t5.z, 22); BQ_BIT(t5.w, 23);
            BQ_BIT(t6.x, 24); BQ_BIT(t6.y, 25); BQ_BIT(t6.z, 26); BQ_BIT(t6.w, 27);
            BQ_BIT(t7.x, 28); BQ_BIT(t7.y, 29); BQ_BIT(t7.z, 30); BQ_BIT(t7.w, 31);

            while (msk) {
                int j = __builtin_ctz(msk);
                msk &= msk - 1;
                float dd = tile[rowbase + j];
                dd = fmaxf(dd, 0.0f);                 // match clip(d2, 0)
                if (cnt < BQ_NSAMPLE || dd < worst) {
                    int pos = (cnt < BQ_NSAMPLE) ? cnt : (BQ_NSAMPLE - 1);
                    while (pos > 0 && dlist[(pos - 1) * 32 + lid] > dd) {
                        dlist[pos * 32 + lid] = dlist[(pos - 1) * 32 + lid];
                        ilist[pos * 32 + lid] = ilist[(pos - 1) * 32 + lid];
                        --pos;
                    }
                    dlist[pos * 32 + lid] = dd;
                    ilist[pos * 32 + lid] = n0 + j;
                    if (cnt < BQ_NSAMPLE) ++cnt;
                    if (cnt == BQ_NSAMPLE) worst = dlist[(BQ_NSAMPLE - 1) * 32 + lid];
                }
            }
        }
        __syncthreads();
    }

    const int q = qbase + lid;
    if (S == 1) {
        // Single-pass: emit directly (pad with nearest index, or 0 if none)
        const int first = (cnt > 0) ? ilist[lid] : 0;
        int* O = out + (size_t)q * BQ_NSAMPLE;
#pragma unroll 1
        for (int j = 0; j < BQ_NSAMPLE; ++j)
            O[j] = (j < cnt) ? ilist[j * 32 + lid] : first;
    } else {
        wc[(size_t)seg * totalQ + q] = cnt;
#pragma unroll 1
        for (int j = 0; j < cnt; ++j) {
            size_t o = ((size_t)(seg * BQ_NSAMPLE + j)) * totalQ + q;  // coalesced over q
            wd[o] = dlist[j * 32 + lid];
            wi[o] = ilist[j * 32 + lid];
        }
    }
}

// S-way merge of per-segment sorted runs; one thread per query.
template <int S>
__global__ __launch_bounds__(256) void ball_query_merge_kernel(
    const float* __restrict__ wd, const int* __restrict__ wi,
    const int* __restrict__ wc, int* __restrict__ out, int totalQ)
{
    const int q = blockIdx.x * 256 + threadIdx.x;
    if (q >= totalQ) return;

    int   head[S], cend[S];
    float curd[S];
    int total = 0;
#pragma unroll
    for (int s = 0; s < S; ++s) {
        head[s] = 0;
        cend[s] = wc[(size_t)s * totalQ + q];
        total  += cend[s];
        curd[s] = (cend[s] > 0) ? wd[((size_t)(s * BQ_NSAMPLE)) * totalQ + q] : 3.0e38f;
    }
    const int outCnt = (total < BQ_NSAMPLE) ? total : BQ_NSAMPLE;
    int firstIdx = 0;
    int* O = out + (size_t)q * BQ_NSAMPLE;

#pragma unroll 1
    for (int j = 0; j < outCnt; ++j) {
        float bestd = curd[0];
        int   bests = 0;
#pragma unroll
        for (int s = 1; s < S; ++s)                 // strict < : ties -> lower segment
            if (curd[s] < bestd) { bestd = curd[s]; bests = s; }
        int idx = 0;
#pragma unroll
        for (int s = 0; s < S; ++s) {
            if (s == bests) {
                idx = wi[((size_t)(s * BQ_NSAMPLE + head[s])) * totalQ + q];
                ++head[s];
                curd[s] = (head[s] < cend[s])
                            ? wd[((size_t)(s * BQ_NSAMPLE + head[s])) * totalQ + q]
                            : 3.0e38f;
            }
        }
        if (j == 0) firstIdx = idx;
        O[j] = idx;
    }
#pragma unroll 1
    for (int j = outCnt; j < BQ_NSAMPLE; ++j) O[j] = firstIdx;
}

extern "C" void kernel_launch(void* const* d_in, const int* in_sizes, int n_in,
                              void* d_out, int out_size, void* d_ws, size_t ws_size,
                              hipStream_t stream) {
    (void)n_in; (void)out_size;
    const float* xyz     = (const float*)d_in[0];
    const float* new_xyz = (const float*)d_in[1];
    int* out = (int*)d_out;

    const int N = in_sizes[0] / (BQ_B * 3);
    const int M = in_sizes[1] / (BQ_B * 3);
    const int totalQ = BQ_B * M;

    // Per-segment workspace: 32 (d2+idx) + 1 count per query  => 260 B * totalQ * S
    const size_t unit = (size_t)totalQ * (BQ_NSAMPLE * 8 + 4);
    int S = 1;
    if (ws_size >= 8 * unit && (N % (8 * 32) == 0)) S = 8;
    else if (ws_size >= 4 * unit && (N % (4 * 32) == 0)) S = 4;
    else if (ws_size >= 2 * unit && (N % (2 * 32) == 0)) S = 2;

    float* wd = (float*)d_ws;
    int*   wi = (int*)((char*)d_ws + (size_t)S * totalQ * BQ_NSAMPLE * 4);
    int*   wc = (int*)((char*)d_ws + (size_t)S * totalQ * BQ_NSAMPLE * 8);

    const int blocks = (totalQ / 32) * S;
    ball_query_seg_kernel<<<blocks, 32, 0, stream>>>(
        xyz, new_xyz, out, wd, wi, wc, N, M, S, totalQ);

    if (S > 1) {
        const int mblocks = (totalQ + 255) / 256;
        if (S == 8)      ball_query_merge_kernel<8><<<mblocks, 256, 0, stream>>>(wd, wi, wc, out, totalQ);
        else if (S == 4) ball_query_merge_kernel<4><<<mblocks, 256, 0, stream>>>(wd, wi, wc, out, totalQ);
        else             ball_query_merge_kernel<2><<<mblocks, 256, 0, stream>>>(wd, wi, wc, out, totalQ);
    }
}